// GNN_33054068310183
// MI455X (gfx1250) — compile-verified
//
#include <hip/hip_runtime.h>
#include <math.h>

// ---------------------------------------------------------------------------
// 2-layer GATConv (PyG-style, concat heads, self loops) for MI455X / gfx1250.
// H = 2 heads, C = 128 channels/head, HC = 256. All fp32.
// GEMMs use native fp32 WMMA (v_wmma_f32_16x16x4_f32); edge softmax/scatter
// are wave32-structured bandwidth kernels.
// ---------------------------------------------------------------------------

typedef __attribute__((ext_vector_type(2))) float v2f;
typedef __attribute__((ext_vector_type(8))) float v8f;

#define TPB 256

// ---------------------------------------------------------------------------
// GEMM: Y[Nrows x 256] = X[Nrows x K] @ W[K x 256], K in {128, 256}.
// One block = 16 output rows x 256 output cols. 8 waves; wave w owns columns
// [32w, 32w+32) as two 16x16 WMMA tiles. A-tile staged in LDS (padded pitch).
// ---------------------------------------------------------------------------
__global__ __launch_bounds__(TPB)
void gat_gemm_wmma(const float* __restrict__ X, const float* __restrict__ W,
                   float* __restrict__ Y, int Nrows, int K) {
    __shared__ float Alds[16 * (256 + 4)];           // up to K=256, padded
    const int pitch = K + 4;
    const int tid   = threadIdx.x;
    const int tileM = blockIdx.x;

    // Stage the 16xK A tile into LDS (coalesced, row by row).
    for (int r = 0; r < 16; ++r) {
        const long grow = (long)tileM * 16 + r;
        const float* src = X + grow * K;
        if (grow < Nrows) {
            for (int c = tid; c < K; c += TPB) Alds[r * pitch + c] = src[c];
        } else {
            for (int c = tid; c < K; c += TPB) Alds[r * pitch + c] = 0.0f;
        }
    }
    __syncthreads();

    const int lane = tid & 31;
    const int wave = tid >> 5;
    const int half = lane >> 4;     // lanes 16-31 hold K+2,K+3 of the frag
    const int l16  = lane & 15;
    const int col0 = wave * 32;     // two 16-wide column tiles: col0, col0+16

    v8f acc0 = {};
    v8f acc1 = {};

    for (int k = 0; k < K; k += 4) {
        const int kk = k + 2 * half;
        // A fragment (16x4 f32): lane l16 -> row M=l16, K = kk, kk+1
        v2f a;
        a.x = Alds[l16 * pitch + kk];
        a.y = Alds[l16 * pitch + kk + 1];
        // B fragments (4x16 f32): lane l16 -> col N=l16, rows kk, kk+1
        const float* wr = W + (long)kk * 256 + col0 + l16;
        v2f b0, b1;
        b0.x = wr[0];        // row kk,   col col0+l16
        b0.y = wr[256];      // row kk+1
        b1.x = wr[16];       // row kk,   col col0+16+l16
        b1.y = wr[256 + 16]; // row kk+1
        acc0 = __builtin_amdgcn_wmma_f32_16x16x4_f32(false, a, false, b0,
                                                     (short)0, acc0, false, false);
        acc1 = __builtin_amdgcn_wmma_f32_16x16x4_f32(false, a, false, b1,
                                                     (short)0, acc1, false, false);
    }

    // D layout: VGPR r, lanes 0-15 -> M=r, lanes 16-31 -> M=r+8, N = l16.
    const int rbase = tileM * 16 + 8 * half;
#pragma unroll
    for (int r = 0; r < 8; ++r) {
        const int row = rbase + r;
        if (row < Nrows) {
            Y[(long)row * 256 + col0 + l16]      = acc0[r];
            Y[(long)row * 256 + col0 + 16 + l16] = acc1[r];
        }
    }
}

// ---------------------------------------------------------------------------
// Per-node attention coefficients: alpha_s[n,h] = xW[n,h,:] . a_src[h,:]
// One wave32 per node; shfl_xor tree reduction.
// ---------------------------------------------------------------------------
__global__ __launch_bounds__(TPB)
void gat_alpha(const float* __restrict__ xW, const float* __restrict__ a_src,
               const float* __restrict__ a_dst, float* __restrict__ as_,
               float* __restrict__ ad_, int Nn) {
    const long gwave = ((long)blockIdx.x * TPB + threadIdx.x) >> 5;
    const int  lane  = threadIdx.x & 31;
    if (gwave >= Nn) return;
    const float* xr = xW + gwave * 256;
    float s0 = 0.f, s1 = 0.f, d0 = 0.f, d1 = 0.f;
#pragma unroll
    for (int c = lane; c < 128; c += 32) {
        const float x0 = xr[c], x1 = xr[128 + c];
        s0 += x0 * a_src[c];       s1 += x1 * a_src[128 + c];
        d0 += x0 * a_dst[c];       d1 += x1 * a_dst[128 + c];
    }
#pragma unroll
    for (int m = 16; m > 0; m >>= 1) {
        s0 += __shfl_xor(s0, m, 32);
        s1 += __shfl_xor(s1, m, 32);
        d0 += __shfl_xor(d0, m, 32);
        d1 += __shfl_xor(d1, m, 32);
    }
    if (lane == 0) {
        as_[gwave * 2 + 0] = s0;  as_[gwave * 2 + 1] = s1;
        ad_[gwave * 2 + 0] = d0;  ad_[gwave * 2 + 1] = d1;
    }
}

// ---------------------------------------------------------------------------
// Helpers
// ---------------------------------------------------------------------------
__global__ void fill_f32(float* __restrict__ p, float v, long n) {
    const long i = (long)blockIdx.x * TPB + threadIdx.x;
    if (i < n) p[i] = v;
}

__device__ __forceinline__ void edge_ends(const int* __restrict__ ei, long E,
                                          long e, int& s, int& d) {
    if (e < E) { s = ei[e]; d = ei[E + e]; }
    else       { s = d = (int)(e - E); }     // self loop
}

__device__ __forceinline__ void atomicMaxFloat(float* addr, float val) {
    unsigned int* ua  = reinterpret_cast<unsigned int*>(addr);
    unsigned int  cur = *ua;
    while (__uint_as_float(cur) < val) {
        const unsigned int assumed = cur;
        cur = atomicCAS(ua, assumed, __float_as_uint(val));
        if (cur == assumed) break;
    }
}

// Pass 1: e = leaky_relu(alpha_s[src] + alpha_d[dst]); segment max at dst.
__global__ __launch_bounds__(TPB)
void gat_edge_logits(const int* __restrict__ ei, long E, long EA,
                     const float* __restrict__ as_, const float* __restrict__ ad_,
                     float* __restrict__ ebuf, float* __restrict__ mbuf) {
    const long e = (long)blockIdx.x * TPB + threadIdx.x;
    if (e >= EA) return;
    int s, d; edge_ends(ei, E, e, s, d);
#pragma unroll
    for (int h = 0; h < 2; ++h) {
        float v = as_[(long)s * 2 + h] + ad_[(long)d * 2 + h];
        v = v > 0.f ? v : 0.2f * v;          // leaky relu, slope 0.2
        ebuf[e * 2 + h] = v;
        atomicMaxFloat(&mbuf[(long)d * 2 + h], v);
    }
}

// Pass 2: p = exp(e - m[dst]); segment sum at dst.
__global__ __launch_bounds__(TPB)
void gat_edge_exp(const int* __restrict__ ei, long E, long EA,
                  const float* __restrict__ mbuf, float* __restrict__ ebuf,
                  float* __restrict__ dbuf) {
    const long e = (long)blockIdx.x * TPB + threadIdx.x;
    if (e >= EA) return;
    int s, d; edge_ends(ei, E, e, s, d);
    (void)s;
#pragma unroll
    for (int h = 0; h < 2; ++h) {
        const float p = expf(ebuf[e * 2 + h] - mbuf[(long)d * 2 + h]);
        ebuf[e * 2 + h] = p;
        atomicAdd(&dbuf[(long)d * 2 + h], p);
    }
}

// Pass 3: out[dst] += (p/denom[dst]) * xW[src].  One wave32 per edge; each
// lane handles 8 coalesced channels -> global_atomic_add_f32.
__global__ __launch_bounds__(TPB)
void gat_edge_scatter(const int* __restrict__ ei, long E, long EA,
                      const float* __restrict__ ebuf, const float* __restrict__ dbuf,
                      const float* __restrict__ xW, float* __restrict__ out) {
    const long gwave = ((long)blockIdx.x * TPB + threadIdx.x) >> 5;
    const int  lane  = threadIdx.x & 31;
    if (gwave >= EA) return;
    int s, d; edge_ends(ei, E, gwave, s, d);
    const float a0 = ebuf[gwave * 2 + 0] / dbuf[(long)d * 2 + 0];
    const float a1 = ebuf[gwave * 2 + 1] / dbuf[(long)d * 2 + 1];
    const float* xr  = xW  + (long)s * 256;
    float*       orow = out + (long)d * 256;
#pragma unroll
    for (int j = 0; j < 8; ++j) {
        const int c = j * 32 + lane;
        const float al = (c < 128) ? a0 : a1;
        atomicAdd(&orow[c], al * xr[c]);
    }
}

// Epilogue: y += bias, optional ELU (alpha = 1).
__global__ __launch_bounds__(TPB)
void gat_bias_act(float* __restrict__ y, const float* __restrict__ b,
                  long n, int hc, int do_elu) {
    const long i = (long)blockIdx.x * TPB + threadIdx.x;
    if (i >= n) return;
    float v = y[i] + b[i % hc];
    if (do_elu) v = v > 0.f ? v : (expf(v) - 1.f);
    y[i] = v;
}

// ---------------------------------------------------------------------------
// Launcher
// ---------------------------------------------------------------------------
static inline unsigned cdivu(long a, long b) { return (unsigned)((a + b - 1) / b); }

extern "C" void kernel_launch(void* const* d_in, const int* in_sizes, int n_in,
                              void* d_out, int out_size, void* d_ws, size_t ws_size,
                              hipStream_t stream) {
    const float* x    = (const float*)d_in[0];
    const int*   ei   = (const int*)d_in[1];
    const float* W1   = (const float*)d_in[2];
    const float* a_s1 = (const float*)d_in[3];
    const float* a_d1 = (const float*)d_in[4];
    const float* b1   = (const float*)d_in[5];
    const float* W2   = (const float*)d_in[6];
    const float* a_s2 = (const float*)d_in[7];
    const float* a_d2 = (const float*)d_in[8];
    const float* b2   = (const float*)d_in[9];
    float* out = (float*)d_out;

    const int  HC  = in_sizes[5];            // 256
    const int  Fin = in_sizes[2] / HC;       // 128
    const int  Nn  = in_sizes[0] / Fin;      // 50000
    const long E   = in_sizes[1] / 2;        // 800000
    const long EA  = E + Nn;                 // edges incl. self loops

    // Workspace layout (floats)
    float* ws    = (float*)d_ws;
    float* bufXW = ws;                        // Nn*256 : xW of current layer
    float* bufH  = bufXW + (long)Nn * 256;    // Nn*256 : layer-1 output h
    float* as_   = bufH  + (long)Nn * 256;    // Nn*2
    float* ad_   = as_   + (long)Nn * 2;      // Nn*2
    float* m_    = ad_   + (long)Nn * 2;      // Nn*2
    float* den_  = m_    + (long)Nn * 2;      // Nn*2
    float* ebuf  = den_  + (long)Nn * 2;      // EA*2

    const long nNH  = (long)Nn * 2;
    const long nNHC = (long)Nn * 256;
    const float NEG_INF = -__builtin_inff();

    // ---------------- Layer 1 ----------------
    gat_gemm_wmma<<<cdivu(Nn, 16), TPB, 0, stream>>>(x, W1, bufXW, Nn, Fin);
    gat_alpha<<<cdivu(Nn, 8), TPB, 0, stream>>>(bufXW, a_s1, a_d1, as_, ad_, Nn);
    fill_f32<<<cdivu(nNH, TPB), TPB, 0, stream>>>(m_, NEG_INF, nNH);
    fill_f32<<<cdivu(nNH, TPB), TPB, 0, stream>>>(den_, 0.f, nNH);
    fill_f32<<<cdivu(nNHC, TPB), TPB, 0, stream>>>(bufH, 0.f, nNHC);
    gat_edge_logits<<<cdivu(EA, TPB), TPB, 0, stream>>>(ei, E, EA, as_, ad_, ebuf, m_);
    gat_edge_exp<<<cdivu(EA, TPB), TPB, 0, stream>>>(ei, E, EA, m_, ebuf, den_);
    gat_edge_scatter<<<cdivu(EA * 32, TPB), TPB, 0, stream>>>(ei, E, EA, ebuf, den_,
                                                              bufXW, bufH);
    gat_bias_act<<<cdivu(nNHC, TPB), TPB, 0, stream>>>(bufH, b1, nNHC, HC, 1);

    // ---------------- Layer 2 ----------------
    gat_gemm_wmma<<<cdivu(Nn, 16), TPB, 0, stream>>>(bufH, W2, bufXW, Nn, HC);
    gat_alpha<<<cdivu(Nn, 8), TPB, 0, stream>>>(bufXW, a_s2, a_d2, as_, ad_, Nn);
    fill_f32<<<cdivu(nNH, TPB), TPB, 0, stream>>>(m_, NEG_INF, nNH);
    fill_f32<<<cdivu(nNH, TPB), TPB, 0, stream>>>(den_, 0.f, nNH);
    fill_f32<<<cdivu(nNHC, TPB), TPB, 0, stream>>>(out, 0.f, nNHC);
    gat_edge_logits<<<cdivu(EA, TPB), TPB, 0, stream>>>(ei, E, EA, as_, ad_, ebuf, m_);
    gat_edge_exp<<<cdivu(EA, TPB), TPB, 0, stream>>>(ei, E, EA, m_, ebuf, den_);
    gat_edge_scatter<<<cdivu(EA * 32, TPB), TPB, 0, stream>>>(ei, E, EA, ebuf, den_,
                                                              bufXW, out);
    gat_bias_act<<<cdivu(nNHC, TPB), TPB, 0, stream>>>(out, b2, nNHC, HC, 0);
}